// HistogramLoss_33981781246065
// MI455X (gfx1250) — compile-verified
//
#include <hip/hip_runtime.h>
#include <math.h>

typedef float v2f __attribute__((ext_vector_type(2)));
typedef float v8f __attribute__((ext_vector_type(8)));

#define N_PTS   8192
#define DIM     128
#define TSIZE   151
#define NCLS    64
#define BT      64            // block tile edge (rows == cols)
#define NTB     (N_PTS / BT)  // 128 tile-blocks per side
#define KCHUNK  32
#define LDS_PAD 2             // 34-word row stride -> conflict-free frag gathers

// ---------------------------------------------------------------------------
// Zero the global histogram (2*151 floats) and class counts (64 ints) in ws.
// Harness poisons ws once and never re-poisons, so every launch must re-init.
// ---------------------------------------------------------------------------
__global__ void init_ws_kernel(float* __restrict__ hist, int* __restrict__ cc) {
    const int t = threadIdx.x;
    if (t < 2 * TSIZE) hist[t] = 0.0f;
    if (t < NCLS) cc[t] = 0;
}

__global__ void class_count_kernel(const int* __restrict__ cls, int* __restrict__ cc) {
    const int i = blockIdx.x * blockDim.x + threadIdx.x;
    if (i < N_PTS) atomicAdd(&cc[cls[i]], 1);
}

// ---------------------------------------------------------------------------
// Fused f32-WMMA GEMM (sims = F * F^T tile) + soft-histogram epilogue.
// One 256-thread block (8 wave32) computes a 64x64 tile of the sims matrix,
// only for upper-triangle tile-blocks, and scatters soft-bin weights into
// wave-private LDS histograms, then reduces to the global histogram.
// ---------------------------------------------------------------------------
__global__ __launch_bounds__(256)
void sim_hist_kernel(const float* __restrict__ F,
                     const int* __restrict__ cls,
                     float* __restrict__ g_hist) {
    const int bi = blockIdx.y;   // row tile-block
    const int bj = blockIdx.x;   // col tile-block
    if (bj < bi) return;         // strict-upper-triangle work only (uniform exit)

    __shared__ float As[BT][KCHUNK + LDS_PAD];
    __shared__ float Bs[BT][KCHUNK + LDS_PAD];
    __shared__ int   clsA[BT], clsB[BT];
    __shared__ float whist[8][2][TSIZE + 1];   // per-wave {pos,neg} histograms

    const int tid  = threadIdx.x;
    const int lane = tid & 31;
    const int w    = tid >> 5;                 // wave id 0..7

    // zero wave-private histograms
    for (int i = tid; i < 8 * 2 * (TSIZE + 1); i += 256)
        ((float*)whist)[i] = 0.0f;
    if (tid < BT) {
        clsA[tid] = cls[bi * BT + tid];
        clsB[tid] = cls[bj * BT + tid];
    }

    // wave w owns C-tiles t0=2w and t0+1 of the 4x4 tile grid (shared tm)
    const int t0   = 2 * w;
    const int tm   = t0 >> 2;                  // tile-row (same for both)
    const int tn0  = t0 & 3;                   // first tile-col
    const int mrow = lane & 15;                // M (or N) within 16x16 tile
    const int kb2  = (lane >> 4) << 1;         // K sub-offset per ISA A/B layout

    v8f acc0 = {};
    v8f acc1 = {};

    for (int kb = 0; kb < DIM; kb += KCHUNK) {
        __syncthreads();
        // cooperative coalesced stage of 64x32 A-rows and B-rows
        #pragma unroll
        for (int e = 0; e < (BT * KCHUNK) / 256; ++e) {
            const int idx = tid + e * 256;
            const int r = idx >> 5;            // /32
            const int c = idx & 31;
            As[r][c] = F[(bi * BT + r) * DIM + kb + c];
            Bs[r][c] = F[(bj * BT + r) * DIM + kb + c];
        }
        __syncthreads();

        #pragma unroll
        for (int ks = 0; ks < KCHUNK; ks += 4) {
            v2f a, b0, b1;
            // A 16x4 f32 frag: lanes 0-15 -> K={0,1}, lanes 16-31 -> K={2,3}
            a.x  = As[tm * 16 + mrow][ks + kb2];
            a.y  = As[tm * 16 + mrow][ks + kb2 + 1];
            // B 4x16 frag for F*F^T: same gather over column-block rows
            b0.x = Bs[tn0 * 16 + mrow][ks + kb2];
            b0.y = Bs[tn0 * 16 + mrow][ks + kb2 + 1];
            b1.x = Bs[(tn0 + 1) * 16 + mrow][ks + kb2];
            b1.y = Bs[(tn0 + 1) * 16 + mrow][ks + kb2 + 1];
            acc0 = __builtin_amdgcn_wmma_f32_16x16x4_f32(
                       false, a, false, b0, (short)0, acc0, false, false);
            acc1 = __builtin_amdgcn_wmma_f32_16x16x4_f32(
                       false, a, false, b1, (short)0, acc1, false, false);
        }
    }

    // ---- epilogue: soft linear binning into wave-private LDS histograms ----
    const float stepv    = 2.0f / (float)(TSIZE - 1);
    const float inv_step = (float)(TSIZE - 1) * 0.5f;
    const int   nidx  = lane & 15;             // N within tile (C layout)
    const int   mhalf = (lane >> 4) << 3;      // +0 or +8 row half (C layout)

    #pragma unroll
    for (int ti = 0; ti < 2; ++ti) {
        const v8f acc = ti ? acc1 : acc0;
        const int tn  = tn0 + ti;
        const int gj  = bj * BT + tn * 16 + nidx;
        const int cj  = clsB[tn * 16 + nidx];
        #pragma unroll
        for (int v = 0; v < 8; ++v) {
            const int m  = v + mhalf;
            const int gi = bi * BT + tm * 16 + m;
            if (gj <= gi) continue;            // strict upper triangle
            const float s = acc[v];
            int idx = (int)floorf((s + 1.0f) * inv_step);
            idx = idx < 0 ? 0 : (idx > TSIZE - 1 ? TSIZE - 1 : idx);
            const float t_idx = -1.0f + (float)idx * stepv;
            const float wlo = (t_idx - s + stepv) * inv_step;
            const float whi = (s - t_idx) * inv_step;
            const int ci = clsA[tm * 16 + m];
            float* h = &whist[w][(ci == cj) ? 0 : 1][0];
            atomicAdd(&h[idx], wlo);
            if (idx + 1 <= TSIZE - 1) atomicAdd(&h[idx + 1], whi);
        }
    }

    __syncthreads();
    // per-block reduction of 8 wave-private histograms -> global atomics
    for (int b = tid; b < TSIZE; b += 256) {
        float sp = 0.0f, sn = 0.0f;
        #pragma unroll
        for (int ww = 0; ww < 8; ++ww) {
            sp += whist[ww][0][b];
            sn += whist[ww][1][b];
        }
        atomicAdd(&g_hist[b], sp);
        atomicAdd(&g_hist[TSIZE + b], sn);
    }
}

// ---------------------------------------------------------------------------
// pos_size = sum_c c*(c-1)/2, neg = N(N-1)/2 - pos; normalize, CDF, dot.
// ---------------------------------------------------------------------------
__global__ void finalize_kernel(const float* __restrict__ g_hist,
                                const int* __restrict__ cc,
                                float* __restrict__ out) {
    if (threadIdx.x == 0) {
        float pos = 0.0f;
        for (int c = 0; c < NCLS; ++c) {
            const float n = (float)cc[c];
            pos += 0.5f * n * (n - 1.0f);
        }
        const float total = 0.5f * (float)N_PTS * (float)(N_PTS - 1);
        const float neg = total - pos;
        float cum = 0.0f, loss = 0.0f;
        for (int b = 0; b < TSIZE; ++b) {
            cum  += g_hist[b] / pos;
            loss += (g_hist[TSIZE + b] / neg) * cum;
        }
        out[0] = loss;
    }
}

extern "C" void kernel_launch(void* const* d_in, const int* in_sizes, int n_in,
                              void* d_out, int out_size, void* d_ws, size_t ws_size,
                              hipStream_t stream) {
    const float* F   = (const float*)d_in[0];
    const int*   cls = (const int*)d_in[1];
    float* hist = (float*)d_ws;                  // [0, 302): pos | neg histograms
    int*   cc   = (int*)(hist + 2 * TSIZE + 2);  // 64 class counts
    float* out  = (float*)d_out;

    init_ws_kernel<<<1, 512, 0, stream>>>(hist, cc);
    class_count_kernel<<<N_PTS / 256, 256, 0, stream>>>(cls, cc);
    sim_hist_kernel<<<dim3(NTB, NTB), 256, 0, stream>>>(F, cls, hist);
    finalize_kernel<<<1, 64, 0, stream>>>(hist, cc, out);
}